// DualMaskRoIPool_39908836114974
// MI455X (gfx1250) — compile-verified
//
#include <hip/hip_runtime.h>
#include <hip/hip_bf16.h>
#include <stdint.h>

// Dual-mask RoI max-pool for MI455X (gfx1250).
// Strategy: block = (1 ROI) x (4 channels). Union-box rows of each channel
// plane are contiguous (row-major 56x56), so we DMA them global->LDS with
// CDNA5 async-tensor path (global_load_async_to_lds_b128, s_wait_asynccnt),
// then 196 threads each compute one (channel, bin) masked max from LDS.

#define FM_C   128
#define FM_H   56
#define FM_W   56
#define NROI   64
#define PH     7
#define PW     7
#define CPB    4      // channels per block
#define THREADS 256

__device__ __forceinline__ void async_ld_b128(uint32_t lds_off, const void* gaddr) {
  // vdst = LDS byte address (low 32 bits of generic LDS pointer),
  // vaddr = 64-bit global address, saddr = off  (GV mode).
  asm volatile("global_load_async_to_lds_b128 %0, %1, off"
               :: "v"(lds_off), "v"((uint64_t)(uintptr_t)gaddr)
               : "memory");
}

__device__ __forceinline__ int clamp_coord(int v, int lim) {
  // Reference clamps only the upper bound; lower clamp is mask-invariant
  // for in-range pixels and only guards memory safety.
  v = (v >= lim) ? (lim - 1) : v;
  return (v < 0) ? 0 : v;
}

__global__ __launch_bounds__(THREADS) void dual_roi_pool(
    const float* __restrict__ feat,    // [128,56,56]
    const float* __restrict__ rois1,   // [64,5]
    const float* __restrict__ rois2,   // [64,5]
    float* __restrict__ out)           // [64,128,7,7]
{
  __shared__ __align__(16) float tile[CPB * FM_H * FM_W];   // 50176 B

  const int c0 = blockIdx.x * CPB;   // channel group base
  const int b  = blockIdx.y;         // ROI index
  const int t  = threadIdx.x;

  // ---- per-ROI geometry (uniform across block -> scalar loads) ----
  const float s = 0.0625f;
  int x1a = clamp_coord((int)rintf(rois1[b*5+1] * s), FM_W);
  int y1a = clamp_coord((int)rintf(rois1[b*5+2] * s), FM_H);
  int x2a = clamp_coord((int)rintf(rois1[b*5+3] * s), FM_W);
  int y2a = clamp_coord((int)rintf(rois1[b*5+4] * s), FM_H);
  int x1b = clamp_coord((int)rintf(rois2[b*5+1] * s), FM_W);
  int y1b = clamp_coord((int)rintf(rois2[b*5+2] * s), FM_H);
  int x2b = clamp_coord((int)rintf(rois2[b*5+3] * s), FM_W);
  int y2b = clamp_coord((int)rintf(rois2[b*5+4] * s), FM_H);

  const int ux1 = min(x1a, x1b), uy1 = min(y1a, y1b);
  const int ux2 = max(x2a, x2b), uy2 = max(y2a, y2b);
  const int h = uy2 - uy1 + 1;       // union box height (>=1)
  const int w = ux2 - ux1 + 1;       // union box width  (>=1)

  // ---- async DMA: rows [uy1..uy2] of CPB planes -> LDS ----
  // Each row = 56 floats = 224 B (14 x b128), 16B-aligned everywhere.
  const int n4 = h * (FM_W / 4);     // float4 transfers per plane
  const uint32_t tile_base = (uint32_t)(uintptr_t)(void*)tile;
  for (int k = 0; k < CPB; ++k) {
    const float4* src =
        (const float4*)(feat + ((size_t)(c0 + k) * FM_H + uy1) * FM_W);
    const uint32_t dst = tile_base + (uint32_t)(((k * FM_H + uy1) * FM_W) * 4);
    for (int i4 = t; i4 < n4; i4 += THREADS)
      async_ld_b128(dst + 16u * (uint32_t)i4, src + i4);
  }
  asm volatile("s_wait_asynccnt 0x0" ::: "memory");
  __syncthreads();

  // ---- masked adaptive max pool: one thread per (channel, bin) ----
  const int nwork = CPB * PH * PW;   // 196
  for (int idx = t; idx < nwork; idx += THREADS) {
    const int k   = idx / (PH * PW);
    const int bin = idx - k * (PH * PW);
    const int i = bin / PW;
    const int j = bin - i * PW;

    const int rs = uy1 + (i * h) / PH;                 // incl
    const int re = uy1 + ((i + 1) * h + PH - 1) / PH;  // excl (>rs always)
    const int cs = ux1 + (j * w) / PW;
    const int ce = ux1 + ((j + 1) * w + PW - 1) / PW;

    const float* plane = tile + k * FM_H * FM_W;
    float m = -1e30f;
    for (int y = rs; y < re; ++y) {
      const bool ay = (y >= y1a) & (y <= y2a);
      const bool by = (y >= y1b) & (y <= y2b);
      const float* row = plane + y * FM_W;
      for (int x = cs; x < ce; ++x) {
        const bool in = (ay & (x >= x1a) & (x <= x2a)) |
                        (by & (x >= x1b) & (x <= x2b));
        const float v = in ? row[x] : 0.0f;   // feat * mask semantics
        m = fmaxf(m, v);
      }
    }
    out[(((size_t)b * FM_C + (c0 + k)) * PH + i) * PW + j] = m;
  }
}

extern "C" void kernel_launch(void* const* d_in, const int* in_sizes, int n_in,
                              void* d_out, int out_size, void* d_ws, size_t ws_size,
                              hipStream_t stream) {
  (void)in_sizes; (void)n_in; (void)out_size; (void)d_ws; (void)ws_size;
  const float* feat  = (const float*)d_in[0];
  const float* rois1 = (const float*)d_in[1];
  const float* rois2 = (const float*)d_in[2];
  float* out = (float*)d_out;

  dim3 grid(FM_C / CPB, NROI);   // (32, 64) blocks
  dual_roi_pool<<<grid, THREADS, 0, stream>>>(feat, rois1, rois2, out);
}